// GCN_30657476559416
// MI455X (gfx1250) — compile-verified
//
#include <hip/hip_runtime.h>

// ---------------------------------------------------------------------------
// GCN forward on MI455X (gfx1250, wave32):
//   3x [ X@W (WMMA f32 16x16x4)  ->  D^-1/2 (A+I) D^-1/2 propagation ]
//   leaky_relu after layer 2, global_add_pool at the end.
// Memory-bound on edge gather/scatter (~2.3 GB total -> ~100us @ 23.3 TB/s);
// GEMMs (9.8 GFLOP fp32) run on the WMMA pipe with exact fp32 accumulation.
// ---------------------------------------------------------------------------

#define N_NODES 100000
#define N_EDGES 640000
#define DIM     128
#define NGRAPH  512

typedef float v2f __attribute__((ext_vector_type(2)));
typedef float v8f __attribute__((ext_vector_type(8)));

// __has_builtin for amdgcn builtins only reports correctly on the device pass;
// on the host pass aux-target builtins are usable but not visible to it.
#if defined(__HIP_DEVICE_COMPILE__)
#if !__has_builtin(__builtin_amdgcn_wmma_f32_16x16x4_f32)
#error "__builtin_amdgcn_wmma_f32_16x16x4_f32 not available on this toolchain (device pass)"
#endif
#endif

// ---------------------------------------------------------------------------
// Y[nrows x 128] = X[nrows x 128] @ W[128 x 128], one wave per 16x16 tile.
// A-frag (16x4 f32): lane<16 -> row M=lane, K=(k0,k0+1); lane>=16 -> K=(k0+2,k0+3)
//   -> a single float2 load per lane per k-step.
// B-frag (4x16 f32): lane<16 -> col N=lane, rows (k0,k0+1); lane>=16 -> rows (k0+2,k0+3).
// C/D (16x16 f32, 8 VGPRs): lane<16 -> rows m0..m0+7; lane>=16 -> rows m0+8..m0+15.
// N_NODES % 16 == 0 and DIM % 16 == 0, so no edge tiles.
// ---------------------------------------------------------------------------
__global__ __launch_bounds__(128) void gemm_wmma_f32(const float* __restrict__ X,
                                                     const float* __restrict__ W,
                                                     float* __restrict__ Y,
                                                     int nrows) {
    const int wave    = (blockIdx.x * blockDim.x + threadIdx.x) >> 5;
    const int lane    = threadIdx.x & 31;
    const int tiles_n = DIM / 16;                 // 8 column tiles
    const int m0      = (wave / tiles_n) * 16;
    const int n0      = (wave % tiles_n) * 16;
    if (m0 >= nrows) return;

    const int half = lane >> 4;                   // 0: K pair (0,1) / rows 0..7
    const int l    = lane & 15;

    v8f acc = {};
    const float* arow = X + (size_t)(m0 + l) * DIM + half * 2;

    #pragma unroll 4
    for (int k = 0; k < DIM; k += 4) {
        v2f a = *(const v2f*)(arow + k);
        const float* bp = W + (size_t)(k + half * 2) * DIM + n0 + l;
        v2f b;
        b.x = bp[0];
        b.y = bp[DIM];
        acc = __builtin_amdgcn_wmma_f32_16x16x4_f32(
            /*neg_a=*/false, a, /*neg_b=*/false, b,
            /*c_mod=*/(short)0, acc, /*reuse_a=*/false, /*reuse_b=*/false);
    }

    float* yrow = Y + (size_t)(m0 + half * 8) * DIM + n0 + l;
    #pragma unroll
    for (int r = 0; r < 8; ++r)
        yrow[(size_t)r * DIM] = acc[r];
}

// ---------------------------------------------------------------------------
// Degree / normalization (computed once; deg includes the self-loop).
// ---------------------------------------------------------------------------
__global__ void deg_init(float* __restrict__ dinv) {
    int i = blockIdx.x * blockDim.x + threadIdx.x;
    if (i < N_NODES) dinv[i] = 1.0f;              // self-loop contribution
}

__global__ void deg_count(const int* __restrict__ col, float* __restrict__ dinv) {
    int e = blockIdx.x * blockDim.x + threadIdx.x;
    if (e < N_EDGES) atomicAdd(&dinv[col[e]], 1.0f);
}

__global__ void deg_finalize(float* __restrict__ dinv) {
    int i = blockIdx.x * blockDim.x + threadIdx.x;
    if (i < N_NODES) {
        float d = dinv[i];
        dinv[i] = (d > 0.0f) ? rsqrtf(d) : 0.0f;
    }
}

// ---------------------------------------------------------------------------
// Self-loop term doubles as output initialization (non-atomic store):
//   out[i,:] = dinv[i]^2 * xw[i,:]
// 32 lanes x float4 covers one 128-wide row.
// ---------------------------------------------------------------------------
__global__ void prop_init(const float* __restrict__ xw,
                          const float* __restrict__ dinv,
                          float* __restrict__ out) {
    int idx = blockIdx.x * blockDim.x + threadIdx.x;   // N_NODES * 32 threads
    int node = idx >> 5;
    if (node >= N_NODES) return;
    int c4 = (idx & 31);
    float s = dinv[node];
    s *= s;
    float4 v = ((const float4*)(xw + (size_t)node * DIM))[c4];
    float4 r = make_float4(s * v.x, s * v.y, s * v.z, s * v.w);
    ((float4*)(out + (size_t)node * DIM))[c4] = r;
}

// ---------------------------------------------------------------------------
// Edge propagation: one wave per edge. float4 gather of xw[row], scaled by
// dinv[row]*dinv[col], global_atomic_add_f32 scatter into out[col].
// ---------------------------------------------------------------------------
__global__ __launch_bounds__(256) void prop_edges(const float* __restrict__ xw,
                                                  const float* __restrict__ dinv,
                                                  const int* __restrict__ row,
                                                  const int* __restrict__ col,
                                                  float* __restrict__ out) {
    int e    = (blockIdx.x * blockDim.x + threadIdx.x) >> 5;
    int lane = threadIdx.x & 31;
    if (e >= N_EDGES) return;
    int r = row[e];
    int c = col[e];
    float norm = dinv[r] * dinv[c];
    float4 v = ((const float4*)(xw + (size_t)r * DIM))[lane];
    float* dst = out + (size_t)c * DIM + lane * 4;
    atomicAdd(dst + 0, norm * v.x);
    atomicAdd(dst + 1, norm * v.y);
    atomicAdd(dst + 2, norm * v.z);
    atomicAdd(dst + 3, norm * v.w);
}

// ---------------------------------------------------------------------------
// Elementwise leaky ReLU (in place on workspace buffer).
// ---------------------------------------------------------------------------
__global__ void leaky_relu(float* __restrict__ h, int n) {
    int i = blockIdx.x * blockDim.x + threadIdx.x;
    if (i < n) {
        float x = h[i];
        h[i] = (x > 0.0f) ? x : 0.01f * x;
    }
}

// ---------------------------------------------------------------------------
// global_add_pool: d_out must be zeroed first (harness poisons it).
// ---------------------------------------------------------------------------
__global__ void pool_zero(float* __restrict__ out) {
    int i = blockIdx.x * blockDim.x + threadIdx.x;
    if (i < NGRAPH * DIM) out[i] = 0.0f;
}

__global__ __launch_bounds__(256) void pool_scatter(const float* __restrict__ h,
                                                    const int* __restrict__ batch,
                                                    float* __restrict__ out) {
    int node = (blockIdx.x * blockDim.x + threadIdx.x) >> 5;
    int lane = threadIdx.x & 31;
    if (node >= N_NODES) return;
    int g = batch[node];
    float4 v = ((const float4*)(h + (size_t)node * DIM))[lane];
    float* dst = out + (size_t)g * DIM + lane * 4;
    atomicAdd(dst + 0, v.x);
    atomicAdd(dst + 1, v.y);
    atomicAdd(dst + 2, v.z);
    atomicAdd(dst + 3, v.w);
}

// ---------------------------------------------------------------------------
// Launcher
// ---------------------------------------------------------------------------
extern "C" void kernel_launch(void* const* d_in, const int* in_sizes, int n_in,
                              void* d_out, int out_size, void* d_ws, size_t ws_size,
                              hipStream_t stream) {
    const float* x   = (const float*)d_in[0];          // [N, D]
    const int*   ei  = (const int*)d_in[1];            // [2, E] flat
    const int*   bat = (const int*)d_in[2];            // [N]
    const float* Wmat[3] = { (const float*)d_in[3],
                             (const float*)d_in[4],
                             (const float*)d_in[5] };  // [D, D] each
    const int* row = ei;                               // edge_index[0]
    const int* col = ei + N_EDGES;                     // edge_index[1]
    float* out = (float*)d_out;                        // [G, D]

    // Workspace layout: dinv [N] | B0 (xw scratch) [N*D] | B1 (h buffer) [N*D]
    float* dinv = (float*)d_ws;
    float* B0   = dinv + ((N_NODES + 255) & ~255);
    float* B1   = B0 + (size_t)N_NODES * DIM;

    const int T = 256;
    dim3 blk(T);

    // Degrees (once; shared by all 3 layers)
    deg_init    <<<(N_NODES + T - 1) / T, blk, 0, stream>>>(dinv);
    deg_count   <<<(N_EDGES + T - 1) / T, blk, 0, stream>>>(col, dinv);
    deg_finalize<<<(N_NODES + T - 1) / T, blk, 0, stream>>>(dinv);

    const int gemm_waves  = (N_NODES / 16) * (DIM / 16);       // 50000
    const int gemm_blocks = (gemm_waves + 3) / 4;              // 4 waves/block
    const int rowT        = (N_NODES * 32 + T - 1) / T;        // wave-per-node grids
    const int edgeBlocks  = (N_EDGES * 32 + T - 1) / T;

    const float* layer_in = x;
    for (int layer = 0; layer < 3; ++layer) {
        gemm_wmma_f32<<<gemm_blocks, dim3(128), 0, stream>>>(layer_in, Wmat[layer], B0, N_NODES);
        prop_init    <<<rowT, blk, 0, stream>>>(B0, dinv, B1);
        prop_edges   <<<edgeBlocks, blk, 0, stream>>>(B0, dinv, row, col, B1);
        if (layer == 1)
            leaky_relu<<<((N_NODES * DIM) + T - 1) / T, blk, 0, stream>>>(B1, N_NODES * DIM);
        layer_in = B1;
    }

    // global_add_pool
    pool_zero   <<<(NGRAPH * DIM + T - 1) / T, blk, 0, stream>>>(out);
    pool_scatter<<<rowT, blk, 0, stream>>>(B1, bat, out);
}